// CoupledGNN_9758165697083
// MI455X (gfx1250) — compile-verified
//
#include <hip/hip_runtime.h>
#include <hip/hip_bf16.h>

// CDNA5 / gfx1250. wave32.
typedef float v2f __attribute__((ext_vector_type(2)));
typedef float v8f __attribute__((ext_vector_type(8)));

#define DCH 32  // feature dim D

__device__ __forceinline__ void atomAddF(float* p, float v) {
    // hardware global_atomic_add_f32 (no CAS loop)
    unsafeAtomicAdd(p, v);
}

// ---------------- zero fill ----------------
__global__ void zero_kernel(float* __restrict__ p, long n) {
    long i = (long)blockIdx.x * blockDim.x + threadIdx.x;
    long stride = (long)gridDim.x * blockDim.x;
    for (; i < n; i += stride) p[i] = 0.0f;
}

// ---------------- instance-norm reduction: per-channel sum / sumsq ----------------
__global__ void norm_reduce_kernel(const float* __restrict__ feat, float* __restrict__ acc, int N) {
    __shared__ float lsum[256];
    __shared__ float lsq[256];
    int tid = threadIdx.x;
    int d = tid & (DCH - 1);
    int rgrp = tid >> 5;  // 0..7
    float sum = 0.f, sq = 0.f;
    for (int n = blockIdx.x * 8 + rgrp; n < N; n += gridDim.x * 8) {
        float x = feat[(size_t)n * DCH + d];
        sum += x;
        sq += x * x;
    }
    lsum[tid] = sum;
    lsq[tid] = sq;
    __syncthreads();
    if (tid < DCH) {
        float a = 0.f, b = 0.f;
        #pragma unroll
        for (int i = 0; i < 8; ++i) { a += lsum[i * DCH + tid]; b += lsq[i * DCH + tid]; }
        atomAddF(&acc[tid], a);
        atomAddF(&acc[DCH + tid], b);
    }
}

// r[b][n][d] = gamma*(x-mean)*rsqrt(var+eps)+beta, replicated across batches
__global__ void norm_apply_kernel(const float* __restrict__ feat, const float* __restrict__ acc,
                                  const float* __restrict__ gamma, const float* __restrict__ beta,
                                  float* __restrict__ r, int N, int B) {
    long i = (long)blockIdx.x * blockDim.x + threadIdx.x;
    long total = (long)N * DCH;
    if (i >= total) return;
    int d = (int)(i & (DCH - 1));
    float invN = 1.0f / (float)N;
    float mean = acc[d] * invN;
    float var = acc[DCH + d] * invN - mean * mean;
    float sc = gamma[d] * rsqrtf(var + 1e-6f);
    float sh = beta[d] - mean * sc;
    float v = feat[i] * sc + sh;
    for (int b = 0; b < B; ++b) r[(long)b * total + i] = v;
}

// ---------------- s init / update / gates / output ----------------
__global__ void s_init_kernel(const float* __restrict__ Xs, const float* __restrict__ act,
                              float* __restrict__ s, int N, int BN) {
    int i = blockIdx.x * blockDim.x + threadIdx.x;
    if (i >= BN) return;
    int n = i % N;
    float x = Xs[i];
    s[i] = (x + act[n]) * (1.0f - x) + x;
}

__global__ void gate_kernel(const float* __restrict__ r, const float* __restrict__ s,
                            const float* __restrict__ gu, const float* __restrict__ gbp,
                            const float* __restrict__ mup, const float* __restrict__ nup,
                            float* __restrict__ t, float* __restrict__ g, int BN) {
    int i = blockIdx.x * blockDim.x + threadIdx.x;
    if (i >= BN) return;
    float gb = gbp[0], mu = mup[0], nu = nup[0];
    const float* rr = r + (size_t)i * DCH;
    float acc = 0.f;
    #pragma unroll
    for (int d = 0; d < DCH; d += 4) {
        float4 rv = *(const float4*)(rr + d);
        float4 gv = *(const float4*)(gu + d);
        acc += rv.x * gv.x + rv.y * gv.y + rv.z * gv.z + rv.w * gv.w;
    }
    t[i] = 1.0f / (1.0f + expf(-(acc + gb)));
    g[i] = mu * s[i] + nu;
}

__global__ void s_update_kernel(const float* __restrict__ Xs, const float* __restrict__ act,
                                const float* __restrict__ msg_s, float* __restrict__ s,
                                int N, int BN) {
    int i = blockIdx.x * blockDim.x + threadIdx.x;
    if (i >= BN) return;
    int n = i % N;
    float x = Xs[i];
    float v = act[n] + msg_s[i];
    v = v > 0.f ? v : 0.f;
    s[i] = v * (1.0f - x) + x;
}

__global__ void final_kernel(const float* __restrict__ Xs, const float* __restrict__ s,
                             float* __restrict__ out, int BN) {
    int i = blockIdx.x * blockDim.x + threadIdx.x;
    if (i >= BN) return;
    float x = Xs[i];
    out[i] = tanhf(s[i]) * (1.0f - x) + x;
}

// ---------------- WMMA fp32 GEMM: OUT[rows,32] = A[rows,32] * W[32,32] ----------------
// One wave computes a 16x32 output tile = two 16x16 C tiles, each via 8x
// v_wmma_f32_16x16x4_f32 (K=4 steps). Exact fp32 math (matches reference dtype).
//
// Layouts (wave32):
//   A 16x4 frag : lane l -> M = l%16, VGPR v -> K = v + 2*(l/16)   [ISA-documented]
//   B 4x16 frag : lane l -> N = l%16, VGPR v -> K = v + 2*(l/16)   [mirror of A/C pattern]
//   C 16x16     : VGPR j, lane l -> M = j + 8*(l/16), N = l%16     [ISA-documented]
template <bool FUSE>
__global__ void gemm32_wmma_kernel(const float* __restrict__ A, const float* __restrict__ W,
                                   const float* __restrict__ MSG, const float* __restrict__ BIAS,
                                   float* __restrict__ OUT, int nTiles) {
    int wave = (int)((blockIdx.x * blockDim.x + threadIdx.x) >> 5);
    int lane = threadIdx.x & 31;
    if (wave >= nTiles) return;  // wave-uniform exit: EXEC stays all-ones for WMMA
    int row0 = wave * 16;
    int m = lane & 15;
    int half = lane >> 4;

    const float* arow = A + (size_t)(row0 + m) * DCH;
    v2f a[8];
    #pragma unroll
    for (int kk = 0; kk < 8; ++kk) {
        float2 p = *(const float2*)(arow + 4 * kk + 2 * half);
        a[kk].x = p.x;
        a[kk].y = p.y;
    }

    #pragma unroll
    for (int nb = 0; nb < 2; ++nb) {
        int col = nb * 16 + m;
        v8f c = {};
        #pragma unroll
        for (int kk = 0; kk < 8; ++kk) {
            v2f bf;
            bf.x = W[(4 * kk + 2 * half + 0) * DCH + col];
            bf.y = W[(4 * kk + 2 * half + 1) * DCH + col];
            c = __builtin_amdgcn_wmma_f32_16x16x4_f32(false, a[kk], false, bf,
                                                      (short)0, c, false, false);
        }
        #pragma unroll
        for (int j = 0; j < 8; ++j) {
            int row = row0 + j + 8 * half;
            size_t idx = (size_t)row * DCH + col;
            float v = c[j];
            if (FUSE) {
                v += MSG[idx] + BIAS[col];
                v = v > 0.f ? v : 0.f;
            }
            OUT[idx] = v;
        }
    }
}

// ---------------- edge scatter: one wave per edge, lane = channel ----------------
__global__ void edge_kernel(const int* __restrict__ esrc, const int* __restrict__ edst,
                            const float* __restrict__ vals, const float* __restrict__ t,
                            const float* __restrict__ g, const float* __restrict__ s,
                            const float* __restrict__ rn, float* __restrict__ msg_r,
                            float* __restrict__ msg_s, int E, int N, int B) {
    int wave = (int)((blockIdx.x * blockDim.x + threadIdx.x) >> 5);
    int lane = threadIdx.x & 31;
    if (wave >= E) return;
    int e = __builtin_amdgcn_readfirstlane(wave);  // force SGPR -> scalar loads of edge data
    int src = esrc[e];
    int dst = edst[e];
    float v = vals[e];
    for (int b = 0; b < B; ++b) {
        long sb = (long)b * N;
        float gg = g[sb + src];  // uniform address -> scalar load
        float rv = rn[(sb + src) * (long)DCH + lane];
        atomAddF(&msg_r[(sb + dst) * (long)DCH + lane], rv * (v * gg));
        if (lane == 0) {
            atomAddF(&msg_s[sb + dst], v * t[sb + src] * s[sb + src]);
        }
    }
}

// ---------------- launcher ----------------
static inline int cdiv(long a, long b) { return (int)((a + b - 1) / b); }

extern "C" void kernel_launch(void* const* d_in, const int* in_sizes, int n_in,
                              void* d_out, int out_size, void* d_ws, size_t ws_size,
                              hipStream_t stream) {
    const float* Xs       = (const float*)d_in[0];   // [B,N,1]
    const float* features = (const float*)d_in[1];   // [N,D]
    const float* self_act = (const float*)d_in[2];   // [N,1]
    const float* values   = (const float*)d_in[3];   // [E]
    const float* gamma    = (const float*)d_in[4];   // [D]
    const float* beta     = (const float*)d_in[5];   // [D]
    const float* W_self   = (const float*)d_in[6];   // [L,D,D]
    const float* W_neigh  = (const float*)d_in[7];   // [L,D,D]
    const float* bvec     = (const float*)d_in[8];   // [L,D]
    const float* gate_u   = (const float*)d_in[9];   // [L,D,1]
    const float* gate_b   = (const float*)d_in[10];  // [L,1]
    const float* mu       = (const float*)d_in[11];  // [L,1]
    const float* nu       = (const float*)d_in[12];  // [L,1]
    const int*   esrc     = (const int*)d_in[13];    // [E]
    const int*   edst     = (const int*)d_in[14];    // [E]
    float* out = (float*)d_out;

    const int N  = in_sizes[1] / DCH;
    const int BN = in_sizes[0];
    const int B  = BN / N;
    const int E  = in_sizes[3];
    const int L  = in_sizes[6] / (DCH * DCH);

    // workspace layout (floats)
    float* w = (float*)d_ws;
    float* acc   = w;  w += 64;
    float* rA    = w;  w += (long)BN * DCH;
    float* rB    = w;  w += (long)BN * DCH;
    float* msg_r = w;  w += (long)BN * DCH;
    float* s     = w;  w += BN;
    float* t     = w;  w += BN;
    float* g     = w;  w += BN;
    float* msg_s = w;  w += BN;

    const int TPB = 256;
    const int nTiles = BN / 16;                 // 100000/16 = 6250, exact
    const int gemmBlocks = cdiv(nTiles, TPB / 32);
    const int edgeBlocks = cdiv(E, TPB / 32);
    const int bnBlocks = cdiv(BN, TPB);

    // 1) instance-norm of features -> rA (both batches)
    zero_kernel<<<1, 64, 0, stream>>>(acc, 64);
    norm_reduce_kernel<<<128, TPB, 0, stream>>>(features, acc, N);
    norm_apply_kernel<<<cdiv((long)N * DCH, TPB), TPB, 0, stream>>>(features, acc, gamma, beta, rA, N, B);

    // 2) s init
    s_init_kernel<<<bnBlocks, TPB, 0, stream>>>(Xs, self_act, s, N, BN);

    float* rcur = rA;
    float* rnxt = rB;
    for (int k = 0; k < L; ++k) {
        // gates from current r,s
        gate_kernel<<<bnBlocks, TPB, 0, stream>>>(rcur, s, gate_u + (long)k * DCH, gate_b + k,
                                                  mu + k, nu + k, t, g, BN);
        // zero message buffers
        zero_kernel<<<4096, TPB, 0, stream>>>(msg_r, (long)BN * DCH);
        zero_kernel<<<256, TPB, 0, stream>>>(msg_s, BN);
        // rn = r @ W_neigh[k]  (WMMA)
        gemm32_wmma_kernel<false><<<gemmBlocks, TPB, 0, stream>>>(
            rcur, W_neigh + (long)k * DCH * DCH, nullptr, nullptr, rnxt, nTiles);
        // edge scatter (both messages)
        edge_kernel<<<edgeBlocks, TPB, 0, stream>>>(esrc, edst, values, t, g, s, rnxt,
                                                    msg_r, msg_s, E, N, B);
        // r_new = relu(r @ W_self[k] + msg_r + b[k])  (fused WMMA), overwrite rn buffer
        gemm32_wmma_kernel<true><<<gemmBlocks, TPB, 0, stream>>>(
            rcur, W_self + (long)k * DCH * DCH, msg_r, bvec + (long)k * DCH, rnxt, nTiles);
        // s update
        s_update_kernel<<<bnBlocks, TPB, 0, stream>>>(Xs, self_act, msg_s, s, N, BN);
        // swap r buffers
        float* tmp = rcur; rcur = rnxt; rnxt = tmp;
    }

    // 3) output
    final_kernel<<<bnBlocks, TPB, 0, stream>>>(Xs, s, out, BN);
}